// STGCN_WOA_PSO_19576460935287
// MI455X (gfx1250) — compile-verified
//
#include <hip/hip_runtime.h>
#include <hip/hip_bf16.h>

typedef __attribute__((ext_vector_type(2))) float v2f;
typedef __attribute__((ext_vector_type(8))) float v8f;

#define N_CH 128

// ---------------- init: deg (self-loop = 1.0), edge-count, cursor ----------------

__global__ void init_k(float* __restrict__ deg, int* __restrict__ cnt,
                       int* __restrict__ cursor, int N) {
    int i = blockIdx.x * blockDim.x + threadIdx.x;
    if (i < N) { deg[i] = 1.0f; cnt[i] = 0; cursor[i] = 0; }
}

// weighted degree + in-degree count per destination node
__global__ void deg_accum_k(const int* __restrict__ colI, const float* __restrict__ w,
                            float* __restrict__ deg, int* __restrict__ cnt, int E) {
    int e = blockIdx.x * blockDim.x + threadIdx.x;
    if (e < E) {
        int c = colI[e];
        atomicAdd(&deg[c], w[e]);
        atomicAdd(&cnt[c], 1);
    }
}

__global__ void dinv_k(const float* __restrict__ deg, float* __restrict__ dinv, int N) {
    int i = blockIdx.x * blockDim.x + threadIdx.x;
    if (i < N) {
        float d = deg[i];
        dinv[i] = d > 0.0f ? rsqrtf(d) : 0.0f;
    }
}

// ---------------- single-block exclusive scan (N up to 1024*items) ----------------

__global__ __launch_bounds__(1024) void exscan_k(const int* __restrict__ cnt,
                                                 int* __restrict__ off, int N) {
    __shared__ int partial[1024];
    const int t = threadIdx.x;
    const int items = (N + 1023) / 1024;
    const int base = t * items;
    int s = 0;
    for (int i = 0; i < items; ++i) {
        int idx = base + i;
        if (idx < N) s += cnt[idx];
    }
    partial[t] = s;
    __syncthreads();
    for (int d = 1; d < 1024; d <<= 1) {          // Hillis-Steele inclusive scan
        int v = (t >= d) ? partial[t - d] : 0;
        __syncthreads();
        partial[t] += v;
        __syncthreads();
    }
    int run = (t == 0) ? 0 : partial[t - 1];      // exclusive base of this chunk
    for (int i = 0; i < items; ++i) {
        int idx = base + i;
        if (idx < N) { off[idx] = run; run += cnt[idx]; }
    }
    if (t == 1023) off[N] = partial[1023];        // total = E
}

// ---------------- CSR fill: pack (row, norm) per incoming edge ----------------

__global__ void csr_fill_k(const int* __restrict__ rowI, const int* __restrict__ colI,
                           const float* __restrict__ w, const float* __restrict__ dinv,
                           const int* __restrict__ off, int* __restrict__ cursor,
                           int2* __restrict__ csr_rn, int E) {
    int e = blockIdx.x * blockDim.x + threadIdx.x;
    if (e >= E) return;
    int c = colI[e], r = rowI[e];
    float nv = dinv[r] * w[e] * dinv[c];
    int p = off[c] + atomicAdd(&cursor[c], 1);
    int2 rn; rn.x = r; rn.y = __float_as_int(nv);
    csr_rn[p] = rn;
}

// ---------------- fp32 WMMA GEMM: Y[N,128] = X[N,128] @ W[128,128]^T ----------------

__global__ __launch_bounds__(256) void gemm128_nt_k(const float* __restrict__ X,
                                                    const float* __restrict__ W,
                                                    float* __restrict__ Y, int N) {
    const int wave = threadIdx.x >> 5;       // col tile 0..7
    const int lane = threadIdx.x & 31;
    const int row0 = blockIdx.x * 16;        // N % 16 == 0 (10000 = 625*16)
    const int col0 = wave * 16;
    const int half = lane >> 4;
    const int l15  = lane & 15;

    v8f acc = {};
#pragma unroll
    for (int k0 = 0; k0 < 128; k0 += 4) {
        const int ka = k0 + half * 2;
        v2f a, b;
        const float* xr = X + (size_t)(row0 + l15) * N_CH + ka;   // A 16x4
        a.x = xr[0]; a.y = xr[1];
        const float* wr = W + (size_t)(col0 + l15) * N_CH + ka;   // B[k][n] = W[n][k]
        b.x = wr[0]; b.y = wr[1];
        acc = __builtin_amdgcn_wmma_f32_16x16x4_f32(false, a, false, b,
                                                    (short)0, acc, false, false);
    }
    const int m0  = half * 8;
    const int col = col0 + l15;
    float* yb = Y + (size_t)row0 * N_CH + col;
#pragma unroll
    for (int r = 0; r < 8; ++r) yb[(size_t)(m0 + r) * N_CH] = acc[r];
}

// ---------------- gather-based GCN aggregation (no float atomics) ----------------
// wave per node; 32 lanes x float4 = 128 channels held in registers.
// h[n] = relu( xw[n]*dinv[n]^2 + sum_in_edges xw[row]*norm + bias )

__global__ __launch_bounds__(256) void gcn_aggregate_k(const float* __restrict__ xw,
                                                       const int2* __restrict__ csr_rn,
                                                       const int* __restrict__ off,
                                                       const float* __restrict__ dinv,
                                                       const float* __restrict__ bias,
                                                       float* __restrict__ hout, int N) {
    int node = blockIdx.x * 8 + (threadIdx.x >> 5);
    if (node >= N) return;
    const int lane = threadIdx.x & 31;
    const int c4 = lane * 4;

    float di = dinv[node];
    float sl = di * di;
    float4 acc = *(const float4*)(xw + (size_t)node * N_CH + c4);   // self-loop msg
    acc.x *= sl; acc.y *= sl; acc.z *= sl; acc.w *= sl;

    const int s = off[node], t = off[node + 1];
    if (s < t) {
        int2 rn = csr_rn[s];                                        // prime pipeline
        for (int j = s; j < t; ++j) {
            const int   r  = rn.x;
            const float nv = __int_as_float(rn.y);
            if (j + 1 < t) rn = csr_rn[j + 1];                      // prefetch next
            const float4 v = *(const float4*)(xw + (size_t)r * N_CH + c4);
            acc.x += v.x * nv; acc.y += v.y * nv;
            acc.z += v.z * nv; acc.w += v.w * nv;
        }
    }
    const float4 b = *(const float4*)(bias + c4);
    acc.x = fmaxf(acc.x + b.x, 0.0f);
    acc.y = fmaxf(acc.y + b.y, 0.0f);
    acc.z = fmaxf(acc.z + b.z, 0.0f);
    acc.w = fmaxf(acc.w + b.w, 0.0f);
    *(float4*)(hout + (size_t)node * N_CH + c4) = acc;
}

// ---------------- temporal conv1d (k=3, SAME) as 3 accumulated WMMA GEMMs ----------------

__global__ __launch_bounds__(256) void conv3_gemm_k(const float* __restrict__ Hin,
                                                    const float* __restrict__ CW,
                                                    const float* __restrict__ CB,
                                                    float* __restrict__ Hout, int N) {
    const int wave = threadIdx.x >> 5;
    const int lane = threadIdx.x & 31;
    const int row0 = blockIdx.x * 16;
    const int col0 = wave * 16;
    const int half = lane >> 4;
    const int l15  = lane & 15;

    v8f acc = {};
    for (int t = 0; t < 3; ++t) {
        const int arow = row0 + l15 + t - 1;
        const int crow = min(max(arow, 0), N - 1);
        const float msk = (arow >= 0 && arow < N) ? 1.0f : 0.0f;   // no divergence at WMMA
        const float* xr = Hin + (size_t)crow * N_CH;
#pragma unroll
        for (int k0 = 0; k0 < 128; k0 += 4) {
            const int ka = k0 + half * 2;
            v2f a, b;
            a.x = xr[ka] * msk;
            a.y = xr[ka + 1] * msk;
            const size_t wb = ((size_t)(col0 + l15) * N_CH + ka) * 3 + t;  // CW[O][I][3]
            b.x = CW[wb];
            b.y = CW[wb + 3];
            acc = __builtin_amdgcn_wmma_f32_16x16x4_f32(false, a, false, b,
                                                        (short)0, acc, false, false);
        }
    }
    const int m0  = half * 8;
    const int col = col0 + l15;
    const float bias = CB[col];
    float* yb = Hout + (size_t)row0 * N_CH + col;
#pragma unroll
    for (int r = 0; r < 8; ++r) yb[(size_t)(m0 + r) * N_CH] = acc[r] + bias;
}

// ---------------- fc head: out[n] = dot(h[n], fc_w) + fc_b ----------------

__global__ __launch_bounds__(256) void fc_k(const float* __restrict__ h,
                                            const float* __restrict__ fw,
                                            const float* __restrict__ fb,
                                            float* __restrict__ out, int N) {
    int node = blockIdx.x * 8 + (threadIdx.x >> 5);
    if (node >= N) return;
    int lane = threadIdx.x & 31;
    const float4 a = *(const float4*)(h + (size_t)node * N_CH + lane * 4);
    const float4 b = *(const float4*)(fw + lane * 4);
    float s = a.x * b.x + a.y * b.y + a.z * b.z + a.w * b.w;
#pragma unroll
    for (int off = 16; off; off >>= 1) s += __shfl_down(s, off, 32);
    if (lane == 0) out[node] = s + fb[0];
}

// ---------------- orchestration ----------------

static inline size_t align4w(size_t w) { return (w + 3) & ~(size_t)3; }  // 16B align (words)

extern "C" void kernel_launch(void* const* d_in, const int* in_sizes, int n_in,
                              void* d_out, int out_size, void* d_ws, size_t ws_size,
                              hipStream_t stream) {
    (void)n_in; (void)out_size; (void)ws_size;
    const float* x       = (const float*)d_in[0];
    const int*   eidx    = (const int*)  d_in[1];   // [2, E] per harness integer ABI
    const float* ew      = (const float*)d_in[2];
    const float* W1      = (const float*)d_in[3];
    const float* b1      = (const float*)d_in[4];
    const float* W2      = (const float*)d_in[5];
    const float* b2      = (const float*)d_in[6];
    const float* conv_w  = (const float*)d_in[7];
    const float* conv_b  = (const float*)d_in[8];
    const float* fc_w    = (const float*)d_in[9];
    const float* fc_b    = (const float*)d_in[10];
    float* out = (float*)d_out;

    const int N  = in_sizes[0] / N_CH;   // 10000
    const int E  = in_sizes[2];          // 640000
    const int NF = N * N_CH;

    // workspace carve-up (word offsets, 16B-aligned where float4/int2 access occurs)
    char* wsb = (char*)d_ws;
    size_t w = 0;
    float* deg    = (float*)(wsb + 4 * w); w = align4w(w + N);
    float* dinv   = (float*)(wsb + 4 * w); w = align4w(w + N);
    int*   cnt    = (int*)  (wsb + 4 * w); w = align4w(w + N);
    int*   cursor = (int*)  (wsb + 4 * w); w = align4w(w + N);
    int*   off    = (int*)  (wsb + 4 * w); w = align4w(w + N + 1);
    int2*  csr_rn = (int2*) (wsb + 4 * w); w = align4w(w + 2 * (size_t)E);
    float* bufA   = (float*)(wsb + 4 * w); w = align4w(w + NF);
    float* buf1   = (float*)(wsb + 4 * w); w = align4w(w + NF);
    float* buf2   = (float*)(wsb + 4 * w); w = align4w(w + NF);

    const int* rowI = eidx;
    const int* colI = eidx + E;

    // gcn_norm + CSR build (gather-form aggregation, no float atomics)
    init_k     <<<(N + 255) / 256, 256, 0, stream>>>(deg, cnt, cursor, N);
    deg_accum_k<<<(E + 255) / 256, 256, 0, stream>>>(colI, ew, deg, cnt, E);
    dinv_k     <<<(N + 255) / 256, 256, 0, stream>>>(deg, dinv, N);
    exscan_k   <<<1, 1024, 0, stream>>>(cnt, off, N);
    csr_fill_k <<<(E + 255) / 256, 256, 0, stream>>>(rowI, colI, ew, dinv, off, cursor,
                                                     csr_rn, E);

    // layer 1: GEMM (WMMA) -> gather aggregate (+bias, relu fused)
    gemm128_nt_k   <<<N / 16, 256, 0, stream>>>(x, W1, bufA, N);
    gcn_aggregate_k<<<(N + 7) / 8, 256, 0, stream>>>(bufA, csr_rn, off, dinv, b1, buf1, N);

    // layer 2
    gemm128_nt_k   <<<N / 16, 256, 0, stream>>>(buf1, W2, bufA, N);
    gcn_aggregate_k<<<(N + 7) / 8, 256, 0, stream>>>(bufA, csr_rn, off, dinv, b2, buf2, N);

    // temporal conv (writes buf1, free now)
    conv3_gemm_k<<<N / 16, 256, 0, stream>>>(buf2, conv_w, conv_b, buf1, N);

    // fc head
    fc_k<<<(N + 7) / 8, 256, 0, stream>>>(buf1, fc_w, fc_b, out, N);
}